// Model_42288247996892
// MI455X (gfx1250) — compile-verified
//
#include <hip/hip_runtime.h>

typedef __attribute__((ext_vector_type(2))) float    v2f;
typedef __attribute__((ext_vector_type(8))) float    v8f;
typedef __attribute__((ext_vector_type(4))) unsigned v4u;
typedef __attribute__((ext_vector_type(8))) int      v8i;
typedef __attribute__((ext_vector_type(4))) int      v4i;

#define BB   16
#define LL   256
#define DD   512
#define NMEL 80
#define TT   3072
#define MTOT (BB * LL)    // 4096 encoder rows
#define M2   (BB * TT)    // 49152 decoder rows
#define PADW 516          // 512 + 2x2-dword TDM pads -> conflict-free LDS banks

__device__ __forceinline__ v8f wmma_f32(v2f a, v2f b, v8f c) {
    return __builtin_amdgcn_wmma_f32_16x16x4_f32(
        false, a, false, b, (short)0, c, false, false);
}

__device__ __forceinline__ unsigned rfl(unsigned x) {
    return (unsigned)__builtin_amdgcn_readfirstlane((int)x);
}

// Tensor Data Mover wrappers: 5-arg builtin on ROCm 7.2, 6-arg on toolchains
// shipping the therock TDM header.
__device__ __forceinline__ void tdm_load(v4u g0, v8i g1, v4i g2, v4i g3) {
#if __has_include(<hip/amd_detail/amd_gfx1250_TDM.h>)
    v8i gz = {};
    __builtin_amdgcn_tensor_load_to_lds(g0, g1, g2, g3, gz, 0);
#else
    __builtin_amdgcn_tensor_load_to_lds(g0, g1, g2, g3, 0);
#endif
}
__device__ __forceinline__ void tdm_store(v4u g0, v8i g1) {
    v4i z4 = {};
#if __has_include(<hip/amd_detail/amd_gfx1250_TDM.h>)
    v8i gz = {};
    __builtin_amdgcn_tensor_store_from_lds(g0, g1, z4, z4, gz, 0);
#else
    __builtin_amdgcn_tensor_store_from_lds(g0, g1, z4, z4, 0);
#endif
}

// -------- Kernel 1: enc = relu((emb[src] + pos) @ W_enc + b_enc) --------
// One wave per 32x16 output tile (2 M sub-tiles share every B fragment).
__global__ __launch_bounds__(256) void enc_kernel(
    const int* __restrict__ src, const float* __restrict__ emb,
    const float* __restrict__ pos, const float* __restrict__ W,
    const float* __restrict__ bias, float* __restrict__ enc)
{
    int wave = (blockIdx.x * blockDim.x + threadIdx.x) >> 5;
    int lane = threadIdx.x & 31;
    int mt = wave >> 5;              // 32-row m-tile [0,128)
    int nt = wave & 31;              // n-tile [0,32)
    int r  = lane & 15;
    int kb = (lane >> 4) << 1;

    int m0 = mt * 32;
    const float* ep0 = emb + (long)src[m0 + r] * DD;
    const float* pp0 = pos + (long)((m0 + r) % LL) * DD;
    const float* ep1 = emb + (long)src[m0 + 16 + r] * DD;
    const float* pp1 = pos + (long)((m0 + 16 + r) % LL) * DD;
    int n = nt * 16 + r;

    v8f c0 = {}, c1 = {};
    for (int k0 = 0; k0 < DD; k0 += 4) {
        v2f b;
        b.x = W[(long)(k0 + kb) * DD + n];
        b.y = W[(long)(k0 + kb + 1) * DD + n];
        v2f a0, a1;
        a0.x = ep0[k0 + kb]     + pp0[k0 + kb];
        a0.y = ep0[k0 + kb + 1] + pp0[k0 + kb + 1];
        a1.x = ep1[k0 + kb]     + pp1[k0 + kb];
        a1.y = ep1[k0 + kb + 1] + pp1[k0 + kb + 1];
        c0 = wmma_f32(a0, b, c0);
        c1 = wmma_f32(a1, b, c1);
    }
    float bv = bias[n];
    int mhi = (lane >> 4) * 8;
#pragma unroll
    for (int rr = 0; rr < 8; ++rr) {
        enc[(long)(m0 + rr + mhi) * DD + n]      = fmaxf(c0[rr] + bv, 0.f);
        enc[(long)(m0 + 16 + rr + mhi) * DD + n] = fmaxf(c1[rr] + bv, 0.f);
    }
}

// -------- Kernel 2: dur_pred = enc @ W_dur + b_dur --------
__global__ __launch_bounds__(256) void dur_kernel(
    const float* __restrict__ enc, const float* __restrict__ Wd,
    const float* __restrict__ bd, float* __restrict__ out)
{
    int wave = (blockIdx.x * blockDim.x + threadIdx.x) >> 5;
    int lane = threadIdx.x & 31;
    const float* rowp = enc + (long)wave * DD;
    float s = 0.f;
    for (int k = lane; k < DD; k += 32) s += rowp[k] * Wd[k];
    for (int off = 16; off; off >>= 1) s += __shfl_xor(s, off, 32);
    if (lane == 0) out[wave] = s + bd[0];
}

// -------- Kernel 3: frame -> token index (cumsum + searchsorted-right) -----
__global__ __launch_bounds__(256) void tok_kernel(
    const int* __restrict__ dur, int* __restrict__ tokbuf)
{
    __shared__ int cs[LL];
    int b = blockIdx.x, tid = threadIdx.x;
    cs[tid] = dur[b * LL + tid];
    __syncthreads();
    for (int off = 1; off < LL; off <<= 1) {
        int v = (tid >= off) ? cs[tid - off] : 0;
        __syncthreads();
        cs[tid] += v;
        __syncthreads();
    }
    for (int t = tid; t < TT; t += 256) {
        int lo = 0, hi = LL;
        while (lo < hi) { int mid = (lo + hi) >> 1; if (cs[mid] <= t) lo = mid + 1; else hi = mid; }
        tokbuf[b * TT + t] = (lo < LL) ? lo : -1;   // -1 => past total duration
    }
}

// -------- Kernel 4: TDM-gather(2x16 rows) -> dec GEMM -> gen GEMM
//          -> one TDM transposed store of an 80x32 mel tile.
// One wave per 32-frame tile (tiles never cross batches: TT % 32 == 0).
__global__ __launch_bounds__(32) void decgen_kernel(
    const float* __restrict__ enc, const int* __restrict__ tokbuf,
    const float* __restrict__ Wdec, const float* __restrict__ bdec,
    const float* __restrict__ Wgen, const float* __restrict__ bgen,
    float* __restrict__ mel)
{
    __shared__ float aT[32 * PADW];     // TDM gather target (padded by TDM)
    __shared__ float decT[32 * PADW];   // relu(dec) tile, same stride
    __shared__ float melT[NMEL * 32];   // [n][t_local] tile for TDM store

    int mt   = blockIdx.x;
    int lane = threadIdx.x & 31;
    int r    = lane & 15;
    int kb   = (lane >> 4) << 1;
    int mhi  = (lane >> 4) * 8;
    int m0   = mt * 32;
    int bA   = m0 / TT;
    int t0   = m0 % TT;

    // ---- pack 32 gather row indices (16-bit, two descriptors of 16).
    // Invalid frames -> index 256 = OOB vs tensor_dim1=256 -> TDM writes
    // zeros. Indices are nondecreasing per descriptor (tok monotone).
    unsigned idx[16];
#pragma unroll
    for (int i = 0; i < 16; ++i) {
        int ta = tokbuf[m0 + 2 * i];
        int tb = tokbuf[m0 + 2 * i + 1];
        unsigned ua = (ta < 0) ? (unsigned)LL : (unsigned)ta;
        unsigned ub = (tb < 0) ? (unsigned)LL : (unsigned)tb;
        idx[i] = rfl(ua | (ub << 16));
    }

    unsigned long long ga = (unsigned long long)(const void*)(enc + (size_t)bA * LL * DD);
    v8i g1;
    g1[0] = 0x03D20000;        // data_size=4B, pad_en, pad_interval=256dw, pad_amount=2dw
    g1[1] = DD << 16;          // tensor_dim0 = 512
    g1[2] = LL << 16;          // tensor_dim1 = 256 rows (index 256 => OOB => zeros)
    g1[3] = DD << 16;          // tile_dim0 = 512
    g1[4] = 16;                // tile_dim1 = 16 gather indices
    g1[5] = DD;                // tensor_dim0_stride = 512
    g1[6] = 0; g1[7] = 0;

    v4u g0;
    g0.x = 0x80000001u;        // count=1, gather_mode, 16-bit indices
    g0.z = rfl((unsigned)ga);
    g0.w = rfl(((unsigned)(ga >> 32) & 0x01FFFFFFu) | (2u << 30));

    v4i g2, g3;
    g0.y = rfl((unsigned)(unsigned long long)(const void*)&aT[0]);
    g2.x = (int)idx[0]; g2.y = (int)idx[1]; g2.z = (int)idx[2]; g2.w = (int)idx[3];
    g3.x = (int)idx[4]; g3.y = (int)idx[5]; g3.z = (int)idx[6]; g3.w = (int)idx[7];
    tdm_load(g0, g1, g2, g3);

    g0.y = rfl((unsigned)(unsigned long long)(const void*)&aT[16 * PADW]);
    g2.x = (int)idx[8];  g2.y = (int)idx[9];  g2.z = (int)idx[10]; g2.w = (int)idx[11];
    g3.x = (int)idx[12]; g3.y = (int)idx[13]; g3.z = (int)idx[14]; g3.w = (int)idx[15];
    tdm_load(g0, g1, g2, g3);

    __builtin_amdgcn_s_wait_tensorcnt(0);
    asm volatile("" ::: "memory");

    // ---- dec = relu(aT @ W_dec + b_dec): each B fragment feeds 2 WMMAs.
    // TDM pad -> row stride PADW, cols >= 256 sit at +2 within the row.
    for (int nc = 0; nc < 32; ++nc) {
        int n = nc * 16 + r;
        v8f c0 = {}, c1 = {};
#pragma unroll
        for (int half = 0; half < 2; ++half) {
            const float* ap0 = &aT[r * PADW + half * 258];
            const float* ap1 = &aT[(r + 16) * PADW + half * 258];
            const float* bp  = &Wdec[(long)(half * 256) * DD];
            for (int kk = 0; kk < 256; kk += 4) {
                v2f b, a0, a1;
                b.x  = bp[(long)(kk + kb) * DD + n];
                b.y  = bp[(long)(kk + kb + 1) * DD + n];
                a0.x = ap0[kk + kb];  a0.y = ap0[kk + kb + 1];
                a1.x = ap1[kk + kb];  a1.y = ap1[kk + kb + 1];
                c0 = wmma_f32(a0, b, c0);
                c1 = wmma_f32(a1, b, c1);
            }
        }
        float bv = bdec[n];
#pragma unroll
        for (int rr = 0; rr < 8; ++rr) {
            decT[(rr + mhi) * PADW + n]        = fmaxf(c0[rr] + bv, 0.f);
            decT[(rr + mhi + 16) * PADW + n]   = fmaxf(c1[rr] + bv, 0.f);
        }
    }

    // ---- mel tile = decT @ W_gen + b_gen, staged [n][t_local 0..31] in LDS
    for (int ng = 0; ng < 5; ++ng) {
        int n = ng * 16 + r;
        v8f c0 = {}, c1 = {};
        for (int k0 = 0; k0 < DD; k0 += 4) {
            v2f b, a0, a1;
            b.x  = Wgen[(long)(k0 + kb) * NMEL + n];
            b.y  = Wgen[(long)(k0 + kb + 1) * NMEL + n];
            a0.x = decT[r * PADW + k0 + kb];
            a0.y = decT[r * PADW + k0 + kb + 1];
            a1.x = decT[(r + 16) * PADW + k0 + kb];
            a1.y = decT[(r + 16) * PADW + k0 + kb + 1];
            c0 = wmma_f32(a0, b, c0);
            c1 = wmma_f32(a1, b, c1);
        }
        float bv = bgen[n];
#pragma unroll
        for (int rr = 0; rr < 8; ++rr) {
            melT[n * 32 + rr + mhi]      = c0[rr] + bv;
            melT[n * 32 + rr + mhi + 16] = c1[rr] + bv;
        }
    }

    // ---- TDM 2D store: 80 rows (stride T) x 32 contiguous frames into
    // mel[bA, :, t0:t0+32]
    asm volatile("s_wait_dscnt 0x0" ::: "memory");  // LDS writes visible to TDM
    unsigned long long gs =
        (unsigned long long)(void*)(mel + (size_t)bA * NMEL * TT + t0);
    v4u s0;
    s0.x = 1u;                                      // count=1, no gather
    s0.y = rfl((unsigned)(unsigned long long)(void*)&melT[0]);
    s0.z = rfl((unsigned)gs);
    s0.w = rfl(((unsigned)(gs >> 32) & 0x01FFFFFFu) | (2u << 30));
    v8i s1;
    s1[0] = 0x00020000;            // data_size = 4B
    s1[1] = (TT & 0xFFFF) << 16;   // tensor_dim0 = 3072
    s1[2] = NMEL << 16;            // tensor_dim1 = 80
    s1[3] = 32 << 16;              // tile_dim0 = 32 (along t, contiguous)
    s1[4] = NMEL;                  // tile_dim1 = 80
    s1[5] = TT;                    // tensor_dim0_stride = 3072
    s1[6] = 0; s1[7] = 0;
    tdm_store(s0, s1);
    __builtin_amdgcn_s_wait_tensorcnt(0);
}

extern "C" void kernel_launch(void* const* d_in, const int* in_sizes, int n_in,
                              void* d_out, int out_size, void* d_ws, size_t ws_size,
                              hipStream_t stream) {
    (void)in_sizes; (void)n_in; (void)out_size; (void)ws_size;
    const int*   src   = (const int*)  d_in[0];
    const int*   dur   = (const int*)  d_in[1];
    /* d_in[2] = T (3072, compile-time constant) */
    const float* emb   = (const float*)d_in[3];
    const float* pos   = (const float*)d_in[4];
    const float* W_enc = (const float*)d_in[5];
    const float* b_enc = (const float*)d_in[6];
    const float* W_dur = (const float*)d_in[7];
    const float* b_dur = (const float*)d_in[8];
    const float* W_dec = (const float*)d_in[9];
    const float* b_dec = (const float*)d_in[10];
    const float* W_gen = (const float*)d_in[11];
    const float* b_gen = (const float*)d_in[12];

    float* mel  = (float*)d_out;                    // [B, NMEL, T]
    float* durp = mel + (long)BB * NMEL * TT;       // [B, L]

    float* enc    = (float*)d_ws;                   // 4096 x 512 f32 (8 MB)
    int*   tokbuf = (int*)(enc + (long)MTOT * DD);  // 49152 int (192 KB)

    // 128 (32-row) m-tiles x 32 n-tiles = 4096 waves, 8 waves/block
    enc_kernel<<<(MTOT / 32) * (DD / 16) / 8, 256, 0, stream>>>(
        src, emb, pos, W_enc, b_enc, enc);
    dur_kernel<<<MTOT / 8, 256, 0, stream>>>(enc, W_dur, b_dur, durp);
    tok_kernel<<<BB, 256, 0, stream>>>(dur, tokbuf);
    // 1536 32-frame tiles, one wave each
    decgen_kernel<<<M2 / 32, 32, 0, stream>>>(
        enc, tokbuf, W_dec, b_dec, W_gen, b_gen, mel);
}